// VN_dgcnn_26525718020209
// MI455X (gfx1250) — compile-verified
//
#include <hip/hip_runtime.h>

// ---------------------------------------------------------------------------
// VN-DGCNN forward for MI455X (gfx1250, wave32).
//  - KNN distance GEMMs + big VN linears use V_WMMA_F32_16X16X4_F32 (exact fp32).
//  - VN leaky-ReLU is fused into the WMMA epilogue fully in-lane (C-tile layout
//    puts all 3 axis components of one (o,n) in the same lane/slot).
//  - Guard-free inner loops: zero-padded A panels / clamped A rows make B-side
//    over-reads harmless (0 * finite = 0; all over-reads stay inside d_ws).
//  - Workspace requirement: ~170 MB of d_ws.
// ---------------------------------------------------------------------------

typedef __attribute__((ext_vector_type(2))) float v2f;
typedef __attribute__((ext_vector_type(8))) float v8f;

#define NPTS   2048
#define BATCH  8
#define KNN    20
#define SLOPE_ 0.2f
#define EPS_   1e-6f
#define CHUNK  512

__device__ __forceinline__ v8f wmma_f32_16x16x4(v2f a, v2f b, v8f c) {
  // (neg_a, A, neg_b, B, c_mod, C, reuse_a, reuse_b)
  return __builtin_amdgcn_wmma_f32_16x16x4_f32(false, a, false, b, (short)0, c,
                                               false, false);
}

// ---------------------------------------------------------------------------
// [B,N,3] -> [B,3,N]
__global__ void transpose_in_kernel(const float* __restrict__ Xin,
                                    float* __restrict__ X0) {
  int t = blockIdx.x * 256 + threadIdx.x;
  if (t >= BATCH * NPTS) return;
  int b = t >> 11, n = t & (NPTS - 1);
  const float* p = Xin + ((long)b * NPTS + n) * 3;
  float* q = X0 + (long)b * 3 * NPTS + n;
  q[0] = p[0];
  q[NPTS] = p[1];
  q[2 * NPTS] = p[2];
}

// sq[b][n] = sum_d X[b][d][n]^2   (X: base + b*sB + d*NPTS + n, d = c*3+axis)
__global__ void sqnorm_kernel(const float* __restrict__ X, long sB, int D,
                              float* __restrict__ sq) {
  int t = blockIdx.x * 256 + threadIdx.x;
  if (t >= BATCH * NPTS) return;
  int b = t >> 11, n = t & (NPTS - 1);
  const float* p = X + (long)b * sB + n;
  float s = 0.0f;
  for (int d = 0; d < D; ++d) {
    float v = p[(long)d * NPTS];
    s += v * v;
  }
  sq[t] = s;
}

// ---------------------------------------------------------------------------
// Fused distance GEMM (WMMA f32 16x16x4) + per-row top-k selection.
// grid: (N/16, B), block: 32 (one wave). Computes neg_dist rows i0..i0+15
// against all columns in CHUNK-sized slabs held in LDS, then 16 lanes keep
// stable sorted top-KNN lists in registers (matches jax.lax.top_k ordering).
// B operand is loaded unguarded up to Dpad: the LDS A panel is zero-padded for
// k >= D, so those products contribute exactly 0 (over-read stays inside ws).
__global__ void __launch_bounds__(32, 1)
dist_topk_kernel(const float* __restrict__ X, long sB, int D, int Dpad,
                 const float* __restrict__ sq, int* __restrict__ knn_idx) {
  __shared__ float rowf[16][132];        // zero-padded row features (Dpad<=128)
  __shared__ float sqrow[16];
  __shared__ float slab[16][CHUNK + 4];  // neg_dist slab

  const int lane = threadIdx.x;
  const int b = blockIdx.y;
  const int i0 = blockIdx.x << 4;
  const int hl = lane >> 4;
  const int l15 = lane & 15;
  const float* Xb = X + (long)b * sB;

  for (int r = 0; r < 16; ++r)
    for (int c = lane; c < Dpad; c += 32)
      rowf[r][c] = (c < D) ? Xb[(long)c * NPTS + (i0 + r)] : 0.0f;
  if (lane < 16) sqrow[lane] = sq[(long)b * NPTS + i0 + lane];
  __syncthreads();

  float bv[KNN];
  int bi[KNN];
#pragma unroll
  for (int q = 0; q < KNN; ++q) { bv[q] = -3.0e38f; bi[q] = 0; }

  for (int ch = 0; ch < NPTS / CHUNK; ++ch) {
    // --- WMMA phase: full wave, EXEC all ones ---
    for (int jt = 0; jt < CHUNK / 16; ++jt) {
      const int j0 = ch * CHUNK + jt * 16;
      const float* colp = Xb + j0 + l15;
      v8f acc = {};
      for (int c = 0; c < Dpad; c += 4) {
        const int ka = c + 2 * hl;  // A/B: vgpr0 -> K=0 (lo half) / K=2 (hi)
        v2f av, bb;
        av.x = rowf[l15][ka];
        av.y = rowf[l15][ka + 1];
        bb.x = colp[(long)ka * NPTS];        // unguarded: A is zero for ka>=D
        bb.y = colp[(long)(ka + 1) * NPTS];
        acc = wmma_f32_16x16x4(av, bb, acc);
      }
      const float sqc = sq[(long)b * NPTS + j0 + l15];
#pragma unroll
      for (int r = 0; r < 8; ++r) {
        const int row = r + 8 * hl;  // C layout: vgpr r -> M=r / M=r+8
        slab[row][jt * 16 + l15] = 2.0f * acc[r] - sqrow[row] - sqc;
      }
    }
    __syncthreads();
    // --- selection phase: lane r owns row r ---
    if (lane < 16) {
      for (int j = 0; j < CHUNK; ++j) {
        const float v = slab[lane][j];
        if (v > bv[KNN - 1]) {
          bv[KNN - 1] = v;
          bi[KNN - 1] = ch * CHUNK + j;
#pragma unroll
          for (int p = KNN - 1; p > 0; --p) {
            if (bv[p] > bv[p - 1]) {  // strict > keeps ties stable (index asc)
              float tv = bv[p]; bv[p] = bv[p - 1]; bv[p - 1] = tv;
              int ti = bi[p]; bi[p] = bi[p - 1]; bi[p - 1] = ti;
            }
          }
        }
      }
    }
    __syncthreads();
  }

  if (lane < 16) {
    int* outp = knn_idx + ((long)b * NPTS + i0 + lane) * KNN;
#pragma unroll
    for (int q = 0; q < KNN; ++q) outp[q] = bi[q];
  }
}

// ---------------------------------------------------------------------------
// Edge conv: per point gather k neighbors, edge feat [diff, center] (2C x 3),
// p = W e, d = Dm e, rejection LeakyReLU, mean over k. grid: B*N, block: 128.
__global__ void __launch_bounds__(128)
edgeconv_kernel(const float* __restrict__ X, long sBx, int Cin, int OUT,
                const int* __restrict__ idxbuf, const float* __restrict__ W,
                const float* __restrict__ Dm, float* __restrict__ Y, long sBy) {
  __shared__ float ctr[42 * 3];
  __shared__ float ef[2 * 42 * 3];
  const int id = blockIdx.x;
  const int b = id >> 11;
  const int n = id & (NPTS - 1);
  const int tid = threadIdx.x;
  const float* Xb = X + (long)b * sBx;
  const int C3 = Cin * 3;
  const int twoC = 2 * Cin;

  for (int t = tid; t < C3; t += blockDim.x) {
    int c = t / 3, ax = t - c * 3;
    float v = Xb[(long)c * (3 * NPTS) + (long)ax * NPTS + n];
    ctr[t] = v;
    ef[C3 + t] = v;  // center half is constant over neighbors
  }
  __syncthreads();

  float a0 = 0.0f, a1 = 0.0f, a2 = 0.0f;
  for (int q = 0; q < KNN; ++q) {
    const int j = idxbuf[((long)b * NPTS + n) * KNN + q];
    for (int t = tid; t < C3; t += blockDim.x) {
      int c = t / 3, ax = t - c * 3;
      ef[t] = Xb[(long)c * (3 * NPTS) + (long)ax * NPTS + j] - ctr[t];
    }
    __syncthreads();
    if (tid < OUT) {
      float p0 = 0, p1 = 0, p2 = 0, d0 = 0, d1 = 0, d2 = 0;
      for (int c2 = 0; c2 < twoC; ++c2) {
        const float w = W[(long)tid * twoC + c2];
        const float dm = Dm[(long)tid * twoC + c2];
        const float e0 = ef[c2 * 3 + 0];
        const float e1 = ef[c2 * 3 + 1];
        const float e2 = ef[c2 * 3 + 2];
        p0 += w * e0; p1 += w * e1; p2 += w * e2;
        d0 += dm * e0; d1 += dm * e1; d2 += dm * e2;
      }
      const float dot = p0 * d0 + p1 * d1 + p2 * d2;
      const float dsq = d0 * d0 + d1 * d1 + d2 * d2;
      const float coef = (dot >= 0.0f) ? 0.0f : dot / (dsq + EPS_);
      a0 += SLOPE_ * p0 + (1.0f - SLOPE_) * (p0 - coef * d0);
      a1 += SLOPE_ * p1 + (1.0f - SLOPE_) * (p1 - coef * d1);
      a2 += SLOPE_ * p2 + (1.0f - SLOPE_) * (p2 - coef * d2);
    }
    __syncthreads();
  }
  if (tid < OUT) {
    const float inv = 1.0f / (float)KNN;
    float* yp = Y + (long)b * sBy + (long)tid * (3 * NPTS) + n;
    yp[0] = a0 * inv;
    yp[NPTS] = a1 * inv;
    yp[2 * NPTS] = a2 * inv;
  }
}

// ---------------------------------------------------------------------------
// WMMA GEMM with fused vector-neuron LeakyReLU epilogue.
// Z[b][o][ax][n] = vn_lrelu over axes of (P = W@X, D = Dm@X).
// X[b][c][ax][n]; optional implicit concat: rows >= Ccut come from XM[b][c'][ax]
// (column-broadcast mean, selected via pointer-select -> single load, no
// branches). dSingle: Dm has a single row broadcast over o. A rows are clamped
// to O-1 (clamped rows only feed never-stored outputs). Main K loop unguarded;
// a guarded tail loop (1-2 iterations) covers the last 3..6 K values.
// grid: (ceil(O/16), N/16, B), block: 32 (one wave).
__global__ void __launch_bounds__(32, 1)
gemm_vn_fused(const float* __restrict__ W, const float* __restrict__ Dm,
              const float* __restrict__ X, long sBx,
              const float* __restrict__ XM, int O, int C, int Ccut,
              float* __restrict__ Z, long sBz, int dSingle) {
  const int lane = threadIdx.x;
  const int hl = lane >> 4;
  const int l15 = lane & 15;
  const int b = blockIdx.z;
  const int o0 = blockIdx.x << 4;
  const int n0 = blockIdx.y << 4;
  const int rowc = min(o0 + l15, O - 1);          // clamped A row (guard-free)
  const long drow = dSingle ? 0L : (long)rowc;
  const int n = n0 + l15;
  const float* Xb = X + (long)b * sBx + (long)n;
  const float* XMb = XM ? (XM + (long)b * 1023) : (const float*)0;
  const float* Wr = W + (long)rowc * C;
  const float* Dr = Dm + drow * C;

  v8f accP[3] = {{}, {}, {}};
  v8f accD[3] = {{}, {}, {}};

  const int Cmain = (C - 3) & ~3;  // main loop: ka+1 <= c+3 < C guaranteed
  int c = 0;
  for (; c < Cmain; c += 4) {
    const int ka = c + 2 * hl;
    v2f aw, ad;
    aw.x = Wr[ka];
    aw.y = Wr[ka + 1];
    ad.x = Dr[ka];
    ad.y = Dr[ka + 1];
    // pointer-select between x5 slab and broadcast mean (no exec divergence)
    const float* p0 = (ka < Ccut) ? (Xb + (long)ka * (3 * NPTS))
                                  : (XMb + (ka - Ccut) * 3);
    const float* p1 = (ka + 1 < Ccut) ? (Xb + (long)(ka + 1) * (3 * NPTS))
                                      : (XMb + (ka + 1 - Ccut) * 3);
    const long s0 = (ka < Ccut) ? (long)NPTS : 1L;     // axis stride per source
    const long s1 = (ka + 1 < Ccut) ? (long)NPTS : 1L;
#pragma unroll
    for (int ax = 0; ax < 3; ++ax) {
      v2f bx;
      bx.x = p0[ax * s0];
      bx.y = p1[ax * s1];
      accP[ax] = wmma_f32_16x16x4(aw, bx, accP[ax]);
      accD[ax] = wmma_f32_16x16x4(ad, bx, accD[ax]);
    }
  }
  // guarded tail (covers remaining 3..6 K values, 1-2 WMMA steps)
  for (; c < C; c += 4) {
    const int ka = c + 2 * hl;
    v2f aw, ad;
    aw.x = (ka < C) ? Wr[ka] : 0.0f;
    aw.y = (ka + 1 < C) ? Wr[ka + 1] : 0.0f;
    ad.x = (ka < C) ? Dr[ka] : 0.0f;
    ad.y = (ka + 1 < C) ? Dr[ka + 1] : 0.0f;
#pragma unroll
    for (int ax = 0; ax < 3; ++ax) {
      float b0 = 0.0f, b1 = 0.0f;
      if (ka < C)
        b0 = (ka < Ccut) ? Xb[(long)ka * (3 * NPTS) + (long)ax * NPTS]
                         : XMb[(ka - Ccut) * 3 + ax];
      if (ka + 1 < C)
        b1 = (ka + 1 < Ccut) ? Xb[(long)(ka + 1) * (3 * NPTS) + (long)ax * NPTS]
                             : XMb[(ka + 1 - Ccut) * 3 + ax];
      v2f bx;
      bx.x = b0;
      bx.y = b1;
      accP[ax] = wmma_f32_16x16x4(aw, bx, accP[ax]);
      accD[ax] = wmma_f32_16x16x4(ad, bx, accD[ax]);
    }
  }

#pragma unroll
  for (int r = 0; r < 8; ++r) {
    const int orow = o0 + r + 8 * hl;
    if (orow < O) {
      const float p0 = accP[0][r], p1 = accP[1][r], p2 = accP[2][r];
      const float d0 = accD[0][r], d1 = accD[1][r], d2 = accD[2][r];
      const float dot = p0 * d0 + p1 * d1 + p2 * d2;
      const float dsq = d0 * d0 + d1 * d1 + d2 * d2;
      const float coef = (dot >= 0.0f) ? 0.0f : dot / (dsq + EPS_);
      float* zp = Z + (long)b * sBz + (long)orow * (3 * NPTS) + n;
      zp[0] = SLOPE_ * p0 + (1.0f - SLOPE_) * (p0 - coef * d0);
      zp[NPTS] = SLOPE_ * p1 + (1.0f - SLOPE_) * (p1 - coef * d1);
      zp[2 * NPTS] = SLOPE_ * p2 + (1.0f - SLOPE_) * (p2 - coef * d2);
    }
  }
}

// ---------------------------------------------------------------------------
// xm[b][c][ax] = mean_n x5[b][c][ax][n]. grid: B*341*3, block: 256.
__global__ void colmean_kernel(const float* __restrict__ X5,
                               float* __restrict__ XM) {
  __shared__ float s[256];
  const int id = blockIdx.x;
  const int b = id / 1023;
  const int rem = id - b * 1023;
  const float* p = X5 + (long)b * (341L * 3 * NPTS) + (long)rem * NPTS;
  float a = 0.0f;
  for (int n = threadIdx.x; n < NPTS; n += 256) a += p[n];
  s[threadIdx.x] = a;
  __syncthreads();
  for (int st = 128; st > 0; st >>= 1) {
    if (threadIdx.x < st) s[threadIdx.x] += s[threadIdx.x + st];
    __syncthreads();
  }
  if (threadIdx.x == 0) XM[id] = s[0] * (1.0f / NPTS);
}

// z0[b][o][ax][n] = sum_c Wlin[o][c] * z2[b][c][ax][n]   (O=3, C=170)
__global__ void wlin_kernel(const float* __restrict__ Z2,
                            const float* __restrict__ Wl,
                            float* __restrict__ Z0) {
  int t = blockIdx.x * 256 + threadIdx.x;
  if (t >= BATCH * 3 * NPTS) return;
  int b = t / (3 * NPTS);
  int rem = t - b * 3 * NPTS;
  int ax = rem / NPTS;
  int n = rem - ax * NPTS;
  const float* z2 = Z2 + (long)b * (170L * 3 * NPTS) + (long)ax * NPTS + n;
  float a0 = 0, a1 = 0, a2 = 0;
  for (int c = 0; c < 170; ++c) {
    float v = z2[(long)c * 3 * NPTS];
    a0 += Wl[c] * v;
    a1 += Wl[170 + c] * v;
    a2 += Wl[340 + c] * v;
  }
  float* z0 = Z0 + (long)b * 9 * NPTS;
  z0[(0 * 3 + ax) * NPTS + n] = a0;
  z0[(1 * 3 + ax) * NPTS + n] = a1;
  z0[(2 * 3 + ax) * NPTS + n] = a2;
}

// ---------------------------------------------------------------------------
// Final stage: xstd[i][k] = sum_j xc2[i][j] * L[k][j] with L = Wlin output;
// out = mean_n vn_lrelu(xstd, W6, D6). Folded: G[o][j] = sum_i W6[o][i]xc2[i][j],
// g[j] = sum_i D6[i]xc2[i][j]; p6 = G Lᵀ, d6 = g Lᵀ. grid: (B, N/256).
__global__ void final_partial_kernel(const float* __restrict__ X5,
                                     const float* __restrict__ XM,
                                     const float* __restrict__ Z0,
                                     const float* __restrict__ W6,
                                     const float* __restrict__ D6,
                                     float* __restrict__ part) {
  __shared__ float s[256];
  const int b = blockIdx.x;
  const int n = blockIdx.y * 256 + threadIdx.x;
  const float* x5b = X5 + (long)b * (341L * 3 * NPTS);
  const float* xmb = XM + (long)b * 1023;

  float G00 = 0, G01 = 0, G02 = 0, G10 = 0, G11 = 0, G12 = 0, G20 = 0,
        G21 = 0, G22 = 0;
  float g0 = 0, g1 = 0, g2 = 0;
  for (int i = 0; i < 682; ++i) {
    float x0, x1, x2;
    if (i < 341) {
      const float* p = x5b + (long)i * 3 * NPTS + n;
      x0 = p[0]; x1 = p[NPTS]; x2 = p[2 * NPTS];
    } else {
      const float* p = xmb + (i - 341) * 3;
      x0 = p[0]; x1 = p[1]; x2 = p[2];
    }
    const float w0 = W6[i], w1 = W6[682 + i], w2 = W6[2 * 682 + i];
    G00 += w0 * x0; G01 += w0 * x1; G02 += w0 * x2;
    G10 += w1 * x0; G11 += w1 * x1; G12 += w1 * x2;
    G20 += w2 * x0; G21 += w2 * x1; G22 += w2 * x2;
    const float dd = D6[i];
    g0 += dd * x0; g1 += dd * x1; g2 += dd * x2;
  }

  float L[3][3];
#pragma unroll
  for (int kk = 0; kk < 3; ++kk)
#pragma unroll
    for (int j = 0; j < 3; ++j)
      L[kk][j] = Z0[(long)b * 9 * NPTS + (kk * 3 + j) * NPTS + n];

  float dv[3];
#pragma unroll
  for (int kk = 0; kk < 3; ++kk)
    dv[kk] = g0 * L[kk][0] + g1 * L[kk][1] + g2 * L[kk][2];
  const float dsq = dv[0] * dv[0] + dv[1] * dv[1] + dv[2] * dv[2];

  const float Gr[3][3] = {{G00, G01, G02}, {G10, G11, G12}, {G20, G21, G22}};
  float h[9];
#pragma unroll
  for (int o = 0; o < 3; ++o) {
    float pv[3];
#pragma unroll
    for (int kk = 0; kk < 3; ++kk)
      pv[kk] = Gr[o][0] * L[kk][0] + Gr[o][1] * L[kk][1] + Gr[o][2] * L[kk][2];
    const float dot = pv[0] * dv[0] + pv[1] * dv[1] + pv[2] * dv[2];
    const float coef = (dot >= 0.0f) ? 0.0f : dot / (dsq + EPS_);
#pragma unroll
    for (int kk = 0; kk < 3; ++kk)
      h[o * 3 + kk] =
          SLOPE_ * pv[kk] + (1.0f - SLOPE_) * (pv[kk] - coef * dv[kk]);
  }

  for (int e = 0; e < 9; ++e) {
    s[threadIdx.x] = h[e];
    __syncthreads();
    for (int st = 128; st > 0; st >>= 1) {
      if (threadIdx.x < st) s[threadIdx.x] += s[threadIdx.x + st];
      __syncthreads();
    }
    if (threadIdx.x == 0)
      part[((long)b * gridDim.y + blockIdx.y) * 9 + e] = s[0];
    __syncthreads();
  }
}

__global__ void final_reduce_kernel(const float* __restrict__ part,
                                    float* __restrict__ out, int nb) {
  int t = blockIdx.x * blockDim.x + threadIdx.x;
  if (t >= BATCH * 9) return;
  int b = t / 9, e = t - b * 9;
  float s = 0.0f;
  for (int q = 0; q < nb; ++q) s += part[((long)b * nb + q) * 9 + e];
  out[t] = s * (1.0f / NPTS);
}

// ---------------------------------------------------------------------------
extern "C" void kernel_launch(void* const* d_in, const int* in_sizes, int n_in,
                              void* d_out, int out_size, void* d_ws,
                              size_t ws_size, hipStream_t stream) {
  (void)in_sizes; (void)n_in; (void)out_size; (void)ws_size;

  const float* xin = (const float*)d_in[0];
  const float* W1 = (const float*)d_in[1];
  const float* Dm1 = (const float*)d_in[2];
  const float* W2 = (const float*)d_in[3];
  const float* Dm2 = (const float*)d_in[4];
  const float* W3 = (const float*)d_in[5];
  const float* Dm3 = (const float*)d_in[6];
  const float* W4 = (const float*)d_in[7];
  const float* Dm4 = (const float*)d_in[8];
  const float* W5 = (const float*)d_in[9];
  const float* Dm5 = (const float*)d_in[10];
  const float* Ws1 = (const float*)d_in[11];
  const float* Ds1 = (const float*)d_in[12];
  const float* Ws2 = (const float*)d_in[13];
  const float* Ds2 = (const float*)d_in[14];
  const float* Wlin = (const float*)d_in[15];
  const float* W6 = (const float*)d_in[16];
  const float* D6 = (const float*)d_in[17];

  const long M3 = 3L * NPTS;  // 6144
  float* ws = (float*)d_ws;
  float* x0 = ws;                                     //  49,152 f
  float* sqp = x0 + (long)BATCH * M3;                 //  16,384 f
  int* idxp = (int*)(sqp + (long)BATCH * NPTS);       // 327,680 i
  float* xc = (float*)(idxp + (long)BATCH * NPTS * KNN);
  const long R4 = 8355840;                            // max(xc 169ch, z2 170ch)
  float* x5 = xc + R4;                                // 16,760,832 f
  float* xm = x5 + (long)BATCH * 341 * M3;            //   8,184 f (+pad)
  float* zb = xm + (long)BATCH * 1023 + 8;            // 16,760,832 f
  float* z0 = zb + (long)BATCH * 341 * M3;            // 147,456 f
  float* part = z0 + (long)BATCH * 9 * NPTS;          //     576 f
  float* z2 = xc;  // reuse xc region (dead after x5 GEMM)

  const long sxc = 169L * M3;

  transpose_in_kernel<<<(BATCH * NPTS + 255) / 256, 256, 0, stream>>>(xin, x0);

  // ---- edge-conv layers
  struct Layer {
    const float* X; long sBx; int Cin; int Dpad;
    const float* W; const float* Dm; int OUT; float* Y;
  } layers[4] = {
      {x0, M3, 1, 4, W1, Dm1, 21, xc + 0},
      {xc + 0, sxc, 21, 64, W2, Dm2, 21, xc + 21L * M3},
      {xc + 21L * M3, sxc, 21, 64, W3, Dm3, 42, xc + 42L * M3},
      {xc + 42L * M3, sxc, 42, 128, W4, Dm4, 85, xc + 84L * M3},
  };
  for (int L = 0; L < 4; ++L) {
    const Layer& ly = layers[L];
    const int D = ly.Cin * 3;
    sqnorm_kernel<<<(BATCH * NPTS + 255) / 256, 256, 0, stream>>>(ly.X, ly.sBx,
                                                                  D, sqp);
    dim3 gd(NPTS / 16, BATCH);
    dist_topk_kernel<<<gd, 32, 0, stream>>>(ly.X, ly.sBx, D, ly.Dpad, sqp,
                                            idxp);
    edgeconv_kernel<<<BATCH * NPTS, 128, 0, stream>>>(
        ly.X, ly.sBx, ly.Cin, ly.OUT, idxp, ly.W, ly.Dm, ly.Y, sxc);
  }

  // ---- x5 = vn_lrelu(xc, W5 [341x169], D5 [1x169])  (broadcast d)
  {
    dim3 g((341 + 15) / 16, NPTS / 16, BATCH);
    gemm_vn_fused<<<g, 32, 0, stream>>>(W5, Dm5, xc, sxc, nullptr, 341, 169,
                                        169, x5, 341L * M3, 1);
  }
  colmean_kernel<<<BATCH * 341 * 3, 256, 0, stream>>>(x5, xm);

  // ---- z = vn_lrelu(concat(x5, mean), Ws1 [341x682], Ds1 [341x682])
  {
    dim3 g((341 + 15) / 16, NPTS / 16, BATCH);
    gemm_vn_fused<<<g, 32, 0, stream>>>(Ws1, Ds1, x5, 341L * M3, xm, 341, 682,
                                        341, zb, 341L * M3, 0);
  }
  // ---- z2 = vn_lrelu(z, Ws2 [170x341], Ds2 [170x341])
  {
    dim3 g((170 + 15) / 16, NPTS / 16, BATCH);
    gemm_vn_fused<<<g, 32, 0, stream>>>(Ws2, Ds2, zb, 341L * M3, nullptr, 170,
                                        341, 341, z2, 170L * M3, 0);
  }
  // ---- z0 = Wlin @ z2  (3x170, pure linear)
  wlin_kernel<<<(BATCH * 3 * NPTS + 255) / 256, 256, 0, stream>>>(z2, Wlin, z0);

  // ---- final: xstd = xc2 · z0ᵀ-frame, vn_lrelu(W6, D6), mean over N
  dim3 gf(BATCH, NPTS / 256);
  final_partial_kernel<<<gf, 256, 0, stream>>>(x5, xm, z0, W6, D6, part);
  final_reduce_kernel<<<1, 128, 0, stream>>>(part, (float*)d_out, NPTS / 256);
}